// GraphConv_12421045420338
// MI455X (gfx1250) — compile-verified
//
#include <hip/hip_runtime.h>

#define N_NODES 5000
#define N_EDGES 40000
#define ROWS 48                    // BATCH * SEQ
#define FIN 64
#define FOUT 64
#define FEAT_PER_NODE (ROWS * FIN) // 3072 floats per node

typedef __attribute__((ext_vector_type(2))) float v2f;
typedef __attribute__((ext_vector_type(8))) float v8f;

// ---------------- phase 0: zero deg+cursor (10000 ints) ----------------
__global__ __launch_bounds__(256) void zero_int_kernel(int* __restrict__ p, int n) {
    int i = blockIdx.x * 256 + threadIdx.x;
    if (i < n) p[i] = 0;
}

// ---------------- phase 1: in-degree histogram ----------------
__global__ __launch_bounds__(256) void count_kernel(const int* __restrict__ dst,
                                                    int* __restrict__ deg) {
    int e = blockIdx.x * 256 + threadIdx.x;
    if (e < N_EDGES) atomicAdd(&deg[dst[e]], 1);
}

// ---------------- phase 2: exclusive scan (single block, 1024 thr) ----------------
__global__ __launch_bounds__(1024) void scan_kernel(const int* __restrict__ deg,
                                                    int* __restrict__ row_off) {
    __shared__ int sh[1024];
    const int PER = 5; // 1024*5 >= 5000
    int tid = threadIdx.x;
    int local_pref[PER];
    int sum = 0;
#pragma unroll
    for (int j = 0; j < PER; ++j) {
        int i = tid * PER + j;
        int v = (i < N_NODES) ? deg[i] : 0;
        local_pref[j] = sum;
        sum += v;
    }
    sh[tid] = sum;
    __syncthreads();
    // Hillis-Steele inclusive scan over per-thread sums
    for (int off = 1; off < 1024; off <<= 1) {
        int v = (tid >= off) ? sh[tid - off] : 0;
        __syncthreads();
        sh[tid] += v;
        __syncthreads();
    }
    int base = (tid == 0) ? 0 : sh[tid - 1];
#pragma unroll
    for (int j = 0; j < PER; ++j) {
        int i = tid * PER + j;
        if (i < N_NODES) row_off[i] = base + local_pref[j];
    }
    if (tid == 1023) row_off[N_NODES] = sh[1023];
}

// ---------------- phase 3: bucket edges by destination ----------------
__global__ __launch_bounds__(256) void bucket_kernel(const int* __restrict__ dst,
                                                     const int* __restrict__ src,
                                                     const int* __restrict__ row_off,
                                                     int* __restrict__ cursor,
                                                     int* __restrict__ sorted_src) {
    int e = blockIdx.x * 256 + threadIdx.x;
    if (e < N_EDGES) {
        int d = dst[e];
        int pos = atomicAdd(&cursor[d], 1);
        sorted_src[row_off[d] + pos] = src[e];
    }
}

// ---------------- phase 4: pull-mode segment mean (no float atomics) ----------------
__global__ __launch_bounds__(256) void aggregate_kernel(const float* __restrict__ feat,
                                                        const int* __restrict__ row_off,
                                                        const int* __restrict__ sorted_src,
                                                        float* __restrict__ meanbuf) {
    int n = blockIdx.x;
    int tid = threadIdx.x;
    int beg = row_off[n];
    int end = row_off[n + 1];
    float a0x = 0.f, a0y = 0.f, a0z = 0.f, a0w = 0.f;
    float a1x = 0.f, a1y = 0.f, a1z = 0.f, a1w = 0.f;
    float a2x = 0.f, a2y = 0.f, a2z = 0.f, a2w = 0.f;
    const float4* f4 = (const float4*)feat;
    for (int e = beg; e < end; ++e) {
        int s = sorted_src[e];
        const float4* fb = f4 + (size_t)s * (FEAT_PER_NODE / 4); // 768 float4
        float4 x0 = fb[tid];
        float4 x1 = fb[tid + 256];
        float4 x2 = fb[tid + 512];
        a0x += x0.x; a0y += x0.y; a0z += x0.z; a0w += x0.w;
        a1x += x1.x; a1y += x1.y; a1z += x1.z; a1w += x1.w;
        a2x += x2.x; a2y += x2.y; a2z += x2.z; a2w += x2.w;
    }
    int cnt = end - beg;
    float inv = 1.0f / (float)(cnt > 0 ? cnt : 1);
    float4* mb = (float4*)meanbuf + (size_t)n * (FEAT_PER_NODE / 4);
    mb[tid]       = make_float4(a0x * inv, a0y * inv, a0z * inv, a0w * inv);
    mb[tid + 256] = make_float4(a1x * inv, a1y * inv, a1z * inv, a1w * inv);
    mb[tid + 512] = make_float4(a2x * inv, a2y * inv, a2z * inv, a2w * inv);
}

// ---------------- phase 5: dual GEMM + concat + relu via f32 WMMA ----------------
// One wave computes one 16x16 tile of either nodes_rep (which=0) or agg (which=1).
// Tiles/node = 2 sources * 3 m-tiles * 4 n-tiles = 24; 5000*24 = 120000 waves =
// 15000 blocks * 8 waves (exact, no tail -> uniform EXEC for WMMA).
__global__ __launch_bounds__(256) void gemm_relu_kernel(const float* __restrict__ feat,
                                                        const float* __restrict__ meanbuf,
                                                        const float* __restrict__ W,
                                                        float* __restrict__ out) {
    __shared__ float wT[FIN * FOUT]; // transposed: wT[n*64+k] = W[k*64+n]
    int tid = threadIdx.x;
    for (int i = tid; i < FIN * FOUT; i += 256) {
        int k = i >> 6;
        int n = i & 63;
        wT[n * 64 + k] = W[i];
    }
    __syncthreads();

    int lane = tid & 31;
    int hi = lane >> 4;   // half-wave select
    int lo = lane & 15;
    int w = blockIdx.x * 8 + (tid >> 5);
    int node = w / 24;
    int t = w - node * 24;
    int which = t / 12;        // 0: nodes_rep, 1: agg_messages
    int r = t - which * 12;
    int mt = r >> 2;           // 0..2  (rows 48 = 3*16)
    int nt = r & 3;            // 0..3  (cols 64 = 4*16)

    const float* A = (which ? meanbuf : feat) + (size_t)node * FEAT_PER_NODE;
    // A-fragment (16x4 f32): lane lo = M row; lanes 0-15 hold K{0,1}, 16-31 hold K{2,3}
    const float* arow = A + (size_t)(mt * 16 + lo) * FIN + 2 * hi;
    // B-fragment (4x16 f32): lane lo = N col; same K split -> contiguous after transpose
    const float* brow = &wT[(nt * 16 + lo) * 64 + 2 * hi];

    v8f c = {0.f, 0.f, 0.f, 0.f, 0.f, 0.f, 0.f, 0.f};
#pragma unroll
    for (int kk = 0; kk < 16; ++kk) {
        v2f a = *(const v2f*)(arow + kk * 4);
        v2f b = *(const v2f*)(brow + kk * 4);
        c = __builtin_amdgcn_wmma_f32_16x16x4_f32(false, a, false, b,
                                                  (short)0, c, false, false);
    }

    // D layout: VGPR j -> row (j + 8*hi), col = lo (within tile); relu + concat store
    float* obase = out + (size_t)node * (ROWS * 128) + (size_t)which * 64 + nt * 16 + lo;
#pragma unroll
    for (int j = 0; j < 8; ++j) {
        int row = mt * 16 + hi * 8 + j;
        float v = c[j];
        obase[(size_t)row * 128] = v > 0.f ? v : 0.f;
    }
}

extern "C" void kernel_launch(void* const* d_in, const int* in_sizes, int n_in,
                              void* d_out, int out_size, void* d_ws, size_t ws_size,
                              hipStream_t stream) {
    (void)in_sizes; (void)n_in; (void)out_size; (void)ws_size;
    const float* feat      = (const float*)d_in[0];
    const float* W         = (const float*)d_in[1];
    const int*   edges_dst = (const int*)d_in[2];
    const int*   edges_src = (const int*)d_in[3];
    float* out = (float*)d_out;

    char* ws = (char*)d_ws;
    // layout: mean (61,440,000 B) | deg (20,000) | cursor (20,000) | row_off (20,004) | sorted_src (160,000)
    float* meanbuf    = (float*)ws;
    int*   deg        = (int*)(ws + 61440000);
    int*   cursor     = (int*)(ws + 61460000);
    int*   row_off    = (int*)(ws + 61480000);
    int*   sorted_src = (int*)(ws + 61500004);

    zero_int_kernel<<<(2 * N_NODES + 255) / 256, 256, 0, stream>>>(deg, 2 * N_NODES);
    count_kernel<<<(N_EDGES + 255) / 256, 256, 0, stream>>>(edges_dst, deg);
    scan_kernel<<<1, 1024, 0, stream>>>(deg, row_off);
    bucket_kernel<<<(N_EDGES + 255) / 256, 256, 0, stream>>>(edges_dst, edges_src,
                                                             row_off, cursor, sorted_src);
    aggregate_kernel<<<N_NODES, 256, 0, stream>>>(feat, row_off, sorted_src, meanbuf);
    gemm_relu_kernel<<<15000, 256, 0, stream>>>(feat, meanbuf, W, out);
}